// SlaterDetSampler_3032246911327
// MI455X (gfx1250) — compile-verified
//
#include <hip/hip_runtime.h>
#include <math.h>

// D = 4096 basis states, NP = 32 electrons, B = 8 chains.
// One workgroup per chain; 1024 threads = 32 waves on one WGP.
// corr_d = P_d (Pk^T Xinv Pk) P_d^T  computed as C = P*M via V_WMMA_F32_16X16X4_F32,
// with unnorm_d = sum_n P[d,n]^2 - C[d,n]*P[d,n] fused into the WMMA epilogue.
// Cross-lane reductions use v_permlane16/v_permlanex16 (VALU) instead of ds_bpermute.

typedef __attribute__((ext_vector_type(2))) float v2f;
typedef __attribute__((ext_vector_type(8))) float v8f;

#define D      4096
#define NP     32
#define BLOCK  1024
#define NWAVES (BLOCK / 32)
#define RPT    (D / BLOCK)      // rows of probs per thread = 4
#define NTILES (D / 16)         // 16-row WMMA tiles = 256

// x + (x lane-permuted by XOR mask within each 16-lane group); pure VALU.
__device__ __forceinline__ float permadd16(float x, int s0, int s1) {
    const int i = __float_as_int(x);
    const int p = __builtin_amdgcn_permlane16(i, i, s0, s1, false, false);
    return x + __int_as_float(p);
}
// XOR-mask selects (nibble i = i^m), applied identically to both 16-lane halves.
#define XSEL1_LO  0x67452301
#define XSEL1_HI  0xEFCDAB89
#define XSEL2_LO  0x54761032
#define XSEL2_HI  0xDCFE98BA
#define XSEL4_LO  0x32107654
#define XSEL4_HI  0xBA98FEDC
#define XSEL8_LO  0xFEDCBA98
#define XSEL8_HI  0x76543210

// reduce within each 16-lane group; result replicated across the group
__device__ __forceinline__ float reduce16(float x) {
    x = permadd16(x, XSEL1_LO, XSEL1_HI);
    x = permadd16(x, XSEL2_LO, XSEL2_HI);
    x = permadd16(x, XSEL4_LO, XSEL4_HI);
    x = permadd16(x, XSEL8_LO, XSEL8_HI);
    return x;
}
// full-wave (32 lane) reduce: 16-lane groups then swap halves (identity select)
__device__ __forceinline__ float reduce32(float x) {
    x = reduce16(x);
    const int i = __float_as_int(x);
    const int p = __builtin_amdgcn_permlanex16(i, i, 0x76543210, 0xFEDCBA98, false, false);
    return x + __int_as_float(p);
}

__global__ __launch_bounds__(BLOCK) void slater_chain_kernel(
    const float* __restrict__ eigfunc,   // (D, D); P = first NP columns of each row
    const float* __restrict__ v,         // (D, D)
    const int*   __restrict__ positions, // (B, NP)
    float*       __restrict__ out)       // (B, NP, D)
{
    __shared__ float sXinv[NP][NP];
    __shared__ float sPk[NP][NP];        // row i = P[Kpad[i], :], zero for i >= k
    __shared__ float sT[NP][NP];
    __shared__ float sM[NP][NP + 1];     // padded stride vs LDS banks
    __shared__ float sUnnorm[D];
    __shared__ float sVec[NP];           // P[pos, :]
    __shared__ float sW[NP];             // Pk @ P_pos
    __shared__ float sUU[NP];            // Xinv @ w
    __shared__ float sRed[NWAVES];
    __shared__ float sScal[2];           // [0] = gg, [1] = sum(p)

    const int b    = blockIdx.x;
    const int tid  = threadIdx.x;
    const int lane = tid & 31;
    const int wave = tid >> 5;
    const int lm   = lane & 15;          // 0..15
    const int lh   = lane >> 4;          // 0..1 (half-wave)

    // ---- init ----
    for (int e = tid; e < NP * NP; e += BLOCK) {
        ((float*)sXinv)[e] = 0.0f;
        ((float*)sPk)[e]   = 0.0f;
    }
    float s_acc[RPT];                    // Jastrow accumulator s, in registers
    #pragma unroll
    for (int j = 0; j < RPT; ++j) s_acc[j] = 0.0f;
    __syncthreads();

    for (int k = 0; k < NP; ++k) {
        const int pos = positions[b * NP + k];

        // ---- small algebra: T = Xinv @ Pk ; M = Pk^T @ T (32x32, one elem/thread) ----
        {
            const int i = tid >> 5, j = tid & 31;
            float acc = 0.0f;
            #pragma unroll
            for (int l = 0; l < NP; ++l) acc += sXinv[i][l] * sPk[l][j];
            sT[i][j] = acc;
        }
        __syncthreads();
        {
            const int a = tid >> 5, j = tid & 31;
            float acc = 0.0f;
            #pragma unroll
            for (int i = 0; i < NP; ++i) acc += sPk[i][a] * sT[i][j];
            sM[a][j] = acc;
        }
        __syncthreads();

        // ---- WMMA: per 16-row tile, C(16x32) = Ptile(16x32) @ M(32x32); fused epilogue ----
        for (int t = wave; t < NTILES; t += NWAVES) {
            const int r0 = t * 16;
            v8f c0 = {};   // columns 0..15
            v8f c1 = {};   // columns 16..31
            const float* Pa = eigfunc + (size_t)(r0 + lm) * D;
            #pragma unroll
            for (int kk = 0; kk < NP; kk += 4) {
                const int ka = kk + 2 * lh;
                v2f a;  a[0]  = Pa[ka];             a[1]  = Pa[ka + 1];
                v2f b0; b0[0] = sM[ka][lm];         b0[1] = sM[ka + 1][lm];
                v2f b1; b1[0] = sM[ka][lm + 16];    b1[1] = sM[ka + 1][lm + 16];
                c0 = __builtin_amdgcn_wmma_f32_16x16x4_f32(
                        false, a, false, b0, (short)0, c0, false, false);
                c1 = __builtin_amdgcn_wmma_f32_16x16x4_f32(
                        false, a, false, b1, (short)0, c1, false, false);
            }
            // unnorm_r = sum_n (P[r,n]^2 - C[r,n]*P[r,n]); VALU butterfly per C-row,
            // keep row (r0 + g + 8*lh) in the lane with lm == g, single store per tile.
            float keep = 0.0f;
            #pragma unroll
            for (int g = 0; g < 8; ++g) {
                const int r = r0 + g + 8 * lh;
                const float p0 = eigfunc[(size_t)r * D + lm];
                const float p1 = eigfunc[(size_t)r * D + lm + 16];
                float part = p0 * p0 + p1 * p1 - c0[g] * p0 - c1[g] * p1;
                part = reduce16(part);          // replicated across the 16-lane group
                keep = (lm == g) ? part : keep; // v_cndmask, no EXEC churn
            }
            if (lm < 8) sUnnorm[r0 + lm + 8 * lh] = keep;
        }
        __syncthreads();

        // ---- probabilities: p = |unnorm/(N-k)| * exp(-0.5 s), normalized ----
        float pj[RPT];
        float lsum = 0.0f;
        const float invRem = 1.0f / (float)(NP - k);
        #pragma unroll
        for (int j = 0; j < RPT; ++j) {
            const int d   = tid + BLOCK * j;
            const float u = sUnnorm[d];
            const float p = fabsf(u * invRem) * __expf(-0.5f * s_acc[j]);
            pj[j] = p;
            lsum += p;
        }
        lsum = reduce32(lsum);
        if (lane == 0) sRed[wave] = lsum;
        __syncthreads();
        if (tid == 0) {
            float tot = 0.0f;
            #pragma unroll
            for (int w = 0; w < NWAVES; ++w) tot += sRed[w];
            sScal[1] = tot;
            sScal[0] = 1.0f / sUnnorm[pos];   // gg
        }
        __syncthreads();
        {
            const float invTot = 1.0f / sScal[1];
            float* op = out + ((size_t)b * NP + k) * D;
            #pragma unroll
            for (int j = 0; j < RPT; ++j) op[tid + BLOCK * j] = pj[j] * invTot;
        }

        // ---- rank-1 Xinv update (Sherman-Morrison extension) ----
        if (tid < NP) sVec[tid] = eigfunc[(size_t)pos * D + tid];   // P[pos,:]
        __syncthreads();
        if (tid < NP) {                                             // w = Pk @ P_pos
            float acc = 0.0f;
            #pragma unroll
            for (int a = 0; a < NP; ++a) acc += sPk[tid][a] * sVec[a];
            sW[tid] = acc;
        }
        __syncthreads();
        if (tid < NP) {                                             // uu = Xinv @ w
            float acc = 0.0f;
            #pragma unroll
            for (int l = 0; l < NP; ++l) acc += sXinv[tid][l] * sW[l];
            sUU[tid] = acc;
        }
        __syncthreads();
        const float gg = sScal[0];
        {
            const int i = tid >> 5, j = tid & 31;                   // Xinv += gg*uu*uu^T
            sXinv[i][j] += gg * sUU[i] * sUU[j];                    // rows/cols>=k: uu==0
        }
        __syncthreads();
        if (tid < NP) {                                             // row/col k, then diag
            const float val = -gg * sUU[tid];
            sXinv[k][tid] = val;
            sXinv[tid][k] = val;
            sPk[k][tid]   = sVec[tid];
        }
        if (tid == 0) sXinv[k][k] = gg;   // same wave as above -> ordered LDS stores

        // ---- s += v[:, pos] (column gather) ----
        #pragma unroll
        for (int j = 0; j < RPT; ++j)
            s_acc[j] += v[(size_t)(tid + BLOCK * j) * D + pos];
        __syncthreads();
    }
}

extern "C" void kernel_launch(void* const* d_in, const int* in_sizes, int n_in,
                              void* d_out, int out_size, void* d_ws, size_t ws_size,
                              hipStream_t stream) {
    const float* eigfunc   = (const float*)d_in[0];
    const float* v         = (const float*)d_in[1];
    const int*   positions = (const int*)d_in[2];
    float*       out       = (float*)d_out;
    const int B = in_sizes[2] / NP;      // 8 chains
    slater_chain_kernel<<<dim3(B), dim3(BLOCK), 0, stream>>>(eigfunc, v, positions, out);
}